// StaticCGM_67465346285680
// MI455X (gfx1250) — compile-verified
//
#include <hip/hip_runtime.h>

// ---------------------------------------------------------------------------
// Static CGM winner-take-all, MI455X (gfx1250).
// Bandwidth-bound: 411 MB traffic -> ~17.7 us floor @ 23.3 TB/s.
// Data path: gfx1250 async global->LDS b128 copies (ASYNCcnt) stage the
// group's channel rows; per-lane read-back (no barrier needed: each lane
// consumes only LDS it requested, ordered by s_wait_asynccnt); NT b128 stores.
// WMMA is structurally inapplicable (argmax-of-4 has no MACs).
// ---------------------------------------------------------------------------

typedef float v4f __attribute__((ext_vector_type(4)));
typedef int   v4i __attribute__((vector_size(16)));   // matches builtin param type

#if defined(__AMDGCN__)
typedef __attribute__((address_space(1))) v4i* gptr_v4i;
typedef __attribute__((address_space(3))) v4i* lptr_v4i;
#endif

#define CGM_C   512
#define CGM_HW  3136      // 56*56
#define CGM_HW4 784       // float4 quads per channel plane
#define CGM_S   4         // MAX_GROUP
#define TILE    256       // quads per block

__global__ __launch_bounds__(256)
void StaticCGM_wta_kernel(const float* __restrict__ x,
                          const int*   __restrict__ groups,
                          float*       __restrict__ out)
{
    __shared__ v4f tile[CGM_S][TILE];   // 16 KB staging tile

    const int tid = threadIdx.x;
    const int p4  = blockIdx.x * TILE + tid;   // quad index in channel plane
    const int g   = blockIdx.y;                // group   (uniform per block)
    const int n   = blockIdx.z;                // sample  (uniform per block)
    if (p4 >= CGM_HW4) return;

    // Group channel table (uniform -> scalar loads, uniform branches).
    int cc[CGM_S];
    #pragma unroll
    for (int s = 0; s < CGM_S; ++s)
        cc[s] = groups[g * CGM_S + s];

    const size_t base_n = (size_t)n * (CGM_C * CGM_HW);

#if defined(__AMDGCN__)
    // ---- Stage the (<=4) channel rows via async global->LDS b128 copies ----
    #pragma unroll
    for (int s = 0; s < CGM_S; ++s) {
        if (cc[s] >= 0) {
            const v4f* src = (const v4f*)(x + base_n + (size_t)cc[s] * CGM_HW) + p4;
            __builtin_amdgcn_global_load_async_to_lds_b128(
                (gptr_v4i)(__UINTPTR_TYPE__)src,
                (lptr_v4i)&tile[s][tid],
                /*imm offset*/0, /*cpol*/0);
        }
    }
    // Wave-level wait: all of *our* async LDS writes have landed.
    __builtin_amdgcn_s_wait_asynccnt(0);
#else
    // Host-pass placeholder (never executed on device).
    #pragma unroll
    for (int s = 0; s < CGM_S; ++s)
        if (cc[s] >= 0)
            tile[s][tid] = *((const v4f*)(x + base_n + (size_t)cc[s] * CGM_HW) + p4);
#endif

    // ---- Read back own slots, per-component argmax (first max wins) ----
    const float NEG = -__builtin_inff();
    v4f v[CGM_S];
    #pragma unroll
    for (int s = 0; s < CGM_S; ++s) {
        if (cc[s] >= 0) v[s] = tile[s][tid];
        else            v[s] = (v4f){NEG, NEG, NEG, NEG};
    }

    int   w0 = 0, w1 = 0, w2 = 0, w3 = 0;
    float b0 = v[0].x, b1 = v[0].y, b2 = v[0].z, b3 = v[0].w;
    #pragma unroll
    for (int s = 1; s < CGM_S; ++s) {
        if (v[s].x > b0) { b0 = v[s].x; w0 = s; }
        if (v[s].y > b1) { b1 = v[s].y; w1 = s; }
        if (v[s].z > b2) { b2 = v[s].z; w2 = s; }
        if (v[s].w > b3) { b3 = v[s].w; w3 = s; }
    }

    // ---- Winner keeps value, losers zeroed; output never re-read -> NT ----
    #pragma unroll
    for (int s = 0; s < CGM_S; ++s) {
        if (cc[s] >= 0) {
            v4f o;
            o.x = (w0 == s) ? v[s].x : 0.0f;
            o.y = (w1 == s) ? v[s].y : 0.0f;
            o.z = (w2 == s) ? v[s].z : 0.0f;
            o.w = (w3 == s) ? v[s].w : 0.0f;
            v4f* p = (v4f*)(out + base_n + (size_t)cc[s] * CGM_HW) + p4;
            __builtin_nontemporal_store(o, p);
        }
    }
}

extern "C" void kernel_launch(void* const* d_in, const int* in_sizes, int n_in,
                              void* d_out, int out_size, void* d_ws, size_t ws_size,
                              hipStream_t stream)
{
    const float* x      = (const float*)d_in[0];
    const int*   groups = (const int*)d_in[1];
    float*       out    = (float*)d_out;

    const int G = in_sizes[1] / CGM_S;                // 147
    const int N = in_sizes[0] / (CGM_C * CGM_HW);     // 32

    dim3 grid((CGM_HW4 + TILE - 1) / TILE,            // 4
              (unsigned)G,                            // 147
              (unsigned)N);                           // 32
    StaticCGM_wta_kernel<<<grid, TILE, 0, stream>>>(x, groups, out);
}